// SwingEnhancement_42511586296329
// MI455X (gfx1250) — compile-verified
//
#include <hip/hip_runtime.h>
#include <stdint.h>

// ---- problem constants (from reference) ----
#define DIMS        768
#define KNBR        10
#define NTOK        (16 * 2048)          // B*S = 32768 tokens
#define ROW_BYTES   (DIMS * 4)           // 3072 B per table row
#define SWING_W     0.3f

// ---- kernel geometry ----
#define NBUF          4                  // async row buffers per wave (pipeline depth)
#define WAVES_PER_BLK 4
#define BLOCK         (WAVES_PER_BLK * 32)
#define LDS_PER_WAVE  (NBUF * ROW_BYTES) // 12 KB
#define DYN_LDS       (WAVES_PER_BLK * LDS_PER_WAVE) // 48 KB / block

typedef float v4f __attribute__((ext_vector_type(4)));

// Issue one 3KB table row (6 x B128 per lane-group) into an LDS buffer via the
// CDNA5 async global->LDS DMA path (ASYNCcnt). The leading s_wait_dscnt 0
// guarantees all prior ds_loads from the recycled buffer have completed before
// the async engine may overwrite it (async LDS writes are unordered vs DS ops).
__device__ __forceinline__ void issue_row(const float* __restrict__ table,
                                          int row, uint32_t lds_byte_base,
                                          int lane) {
    uint64_t g = (uint64_t)(uintptr_t)table
               + (uint64_t)(uint32_t)row * (uint64_t)ROW_BYTES
               + (uint32_t)(lane * 16);
    uint32_t l = lds_byte_base + (uint32_t)(lane * 16);
    asm volatile(
        "s_wait_dscnt 0\n\t"
        "global_load_async_to_lds_b128 %0, %1, off\n\t"
        "global_load_async_to_lds_b128 %0, %1, off offset:512\n\t"
        "global_load_async_to_lds_b128 %0, %1, off offset:1024\n\t"
        "global_load_async_to_lds_b128 %0, %1, off offset:1536\n\t"
        "global_load_async_to_lds_b128 %0, %1, off offset:2048\n\t"
        "global_load_async_to_lds_b128 %0, %1, off offset:2560"
        :: "v"(l), "v"(g) : "memory");
}

template <int W>
__device__ __forceinline__ void wait_asynccnt() {
    asm volatile("s_wait_asynccnt %0" :: "n"(W) : "memory");
}

// Wait until row k has landed (in-order ASYNCcnt completion), then accumulate
// it from LDS (ds_load_b128) into the per-lane accumulators.
template <int W>
__device__ __forceinline__ void consume_row(const float* __restrict__ smem_wave,
                                            int k, float wk, int lane,
                                            v4f acc[6]) {
    wait_asynccnt<W>();
    const v4f* b4 =
        reinterpret_cast<const v4f*>(smem_wave + (size_t)(k & (NBUF - 1)) * DIMS);
#pragma unroll
    for (int j = 0; j < 6; ++j) {
        v4f v = b4[j * 32 + lane];   // ds_load_b128
        acc[j] += wk * v;            // contracts to v_fmac per component
    }
}

__global__ __launch_bounds__(BLOCK) void
SwingEnhancement_42511586296329_kernel(const float* __restrict__ x,
                                       const int* __restrict__ item_ids,
                                       const float* __restrict__ item_table,
                                       const int* __restrict__ nbr_idx,
                                       const float* __restrict__ nbr_w,
                                       float* __restrict__ out) {
    extern __shared__ float smem[];

    const int lane  = (int)(threadIdx.x & 31u);
    const int wave  = (int)(threadIdx.x >> 5u);
    const int token = (int)blockIdx.x * WAVES_PER_BLK + wave; // grid exactly covers NTOK

    // LDS byte base of this wave's buffers (dynamic LDS starts at static size).
    const uint32_t lds_base =
        __builtin_amdgcn_groupstaticsize() + (uint32_t)wave * LDS_PER_WAVE;
    const float* smem_wave = smem + (size_t)wave * (NBUF * DIMS);

    // Wave-uniform gather of this token's neighbor ids + weights.
    const int tid = item_ids[token];
    int   ridx[KNBR];
    float rw[KNBR];
#pragma unroll
    for (int k = 0; k < KNBR; ++k) {
        ridx[k] = nbr_idx[(size_t)tid * KNBR + k];
        rw[k]   = nbr_w[(size_t)tid * KNBR + k];
    }

    // Hoist the streamed x tile (LOADcnt) so its HBM latency overlaps the whole
    // async gather pipeline; nontemporal so it doesn't evict table rows from L2.
    const v4f* x4 = reinterpret_cast<const v4f*>(x + (size_t)token * DIMS);
    v4f xv[6];
#pragma unroll
    for (int j = 0; j < 6; ++j)
        xv[j] = __builtin_nontemporal_load(&x4[j * 32 + lane]);

    // ---- software pipeline: 4 rows (24 async B128, 12KB) in flight ----
#pragma unroll
    for (int k = 0; k < NBUF; ++k)
        issue_row(item_table, ridx[k], lds_base + (uint32_t)k * ROW_BYTES, lane);

    v4f acc[6] = {};

    consume_row<18>(smem_wave, 0, rw[0], lane, acc);
    issue_row(item_table, ridx[4], lds_base + ((4) & (NBUF - 1)) * ROW_BYTES, lane);
    consume_row<18>(smem_wave, 1, rw[1], lane, acc);
    issue_row(item_table, ridx[5], lds_base + ((5) & (NBUF - 1)) * ROW_BYTES, lane);
    consume_row<18>(smem_wave, 2, rw[2], lane, acc);
    issue_row(item_table, ridx[6], lds_base + ((6) & (NBUF - 1)) * ROW_BYTES, lane);
    consume_row<18>(smem_wave, 3, rw[3], lane, acc);
    issue_row(item_table, ridx[7], lds_base + ((7) & (NBUF - 1)) * ROW_BYTES, lane);
    consume_row<18>(smem_wave, 4, rw[4], lane, acc);
    issue_row(item_table, ridx[8], lds_base + ((8) & (NBUF - 1)) * ROW_BYTES, lane);
    consume_row<18>(smem_wave, 5, rw[5], lane, acc);
    issue_row(item_table, ridx[9], lds_base + ((9) & (NBUF - 1)) * ROW_BYTES, lane);
    consume_row<18>(smem_wave, 6, rw[6], lane, acc);
    consume_row<12>(smem_wave, 7, rw[7], lane, acc);
    consume_row<6 >(smem_wave, 8, rw[8], lane, acc);
    consume_row<0 >(smem_wave, 9, rw[9], lane, acc);

    // ---- epilogue: out = x + 0.3 * agg  (streamed nontemporal store) ----
    v4f* o4 = reinterpret_cast<v4f*>(out + (size_t)token * DIMS);
#pragma unroll
    for (int j = 0; j < 6; ++j) {
        v4f r = xv[j] + SWING_W * acc[j];
        __builtin_nontemporal_store(r, &o4[j * 32 + lane]);
    }
}

extern "C" void kernel_launch(void* const* d_in, const int* in_sizes, int n_in,
                              void* d_out, int out_size, void* d_ws, size_t ws_size,
                              hipStream_t stream) {
    (void)in_sizes; (void)n_in; (void)out_size; (void)d_ws; (void)ws_size;
    const float* x          = (const float*)d_in[0];
    const int*   item_ids   = (const int*)d_in[1];
    const float* item_table = (const float*)d_in[2];
    const int*   nbr_idx    = (const int*)d_in[3];
    const float* nbr_w      = (const float*)d_in[4];
    float*       out        = (float*)d_out;

    dim3 grid(NTOK / WAVES_PER_BLK);   // 8192 blocks, one token per wave
    dim3 block(BLOCK);                 // 128 threads = 4 wave32
    SwingEnhancement_42511586296329_kernel<<<grid, block, DYN_LDS, stream>>>(
        x, item_ids, item_table, nbr_idx, nbr_w, out);
}